// SMSoftmax_12429635354922
// MI455X (gfx1250) — compile-verified
//
#include <hip/hip_runtime.h>
#include <math.h>

#define EPS_F 1e-8f
#define BLK 512
#define NWAVES (BLK / 32)
#define COLS 32000

// Native clang vector types (HIP's float4 class is rejected by the builtins).
typedef float v4f __attribute__((ext_vector_type(4)));
typedef int   v4i __attribute__((ext_vector_type(4)));

// Address-space-qualified pointers for the gfx1250 async-to-LDS builtin:
// param0 = v4i in global (AS1, printed as __device__), param1 = v4i in LDS (AS3).
typedef __attribute__((address_space(1))) v4i* gas1_v4i;
typedef __attribute__((address_space(3))) v4i* as3_v4i;

#if defined(__HIP_DEVICE_COMPILE__) && __has_builtin(__builtin_amdgcn_global_load_async_to_lds_b128)
#define USE_ASYNC_LDS 1
#else
#define USE_ASYNC_LDS 0
#endif

__device__ __forceinline__ float wave_reduce_max(float v) {
#pragma unroll
  for (int o = 16; o > 0; o >>= 1) v = fmaxf(v, __shfl_xor(v, o, 32));
  return v;
}

__device__ __forceinline__ float wave_reduce_sum(float v) {
#pragma unroll
  for (int o = 16; o > 0; o >>= 1) v += __shfl_xor(v, o, 32);
  return v;
}

// One workgroup per row. Row (128 KB) staged into LDS via async copy, read
// from HBM exactly once; exp computed exactly once per element (LDS row is
// overwritten in-place with e); output written once with non-temporal stores.
__global__ __launch_bounds__(BLK) void sm_softmax_kernel(
    const float* __restrict__ logits,
    const float* __restrict__ dist,
    float* __restrict__ out,
    int C) {
  extern __shared__ float srow[];          // C floats (dynamic LDS)
  __shared__ float s_part[NWAVES];
  __shared__ float s_bval;

  const int tid  = threadIdx.x;
  const int wid  = tid >> 5;
  const int lane = tid & 31;
  const long long row = blockIdx.x;
  const int C4 = C >> 2;

  const float kshift = expf(-dist[0]);     // ed = e * exp(-dist)

  const v4f* grow = (const v4f*)(logits + row * (long long)C);
  v4f*       lrow = (v4f*)srow;
  v4f*       orow = (v4f*)(out + row * (long long)C);

  // ---- Stage: global -> LDS (async, per-lane b128) ----
  for (int i = tid; i < C4; i += BLK) {
#if USE_ASYNC_LDS
    __builtin_amdgcn_global_load_async_to_lds_b128(
        (gas1_v4i)(void*)(grow + i), (as3_v4i)(lrow + i), 0, 0);
#else
    lrow[i] = grow[i];
#endif
  }
#if USE_ASYNC_LDS
#if __has_builtin(__builtin_amdgcn_s_wait_asynccnt)
  __builtin_amdgcn_s_wait_asynccnt(0);
#else
  asm volatile("s_wait_asynccnt 0x0" ::: "memory");
#endif
#endif
  __syncthreads();

  // ---- Pass A: row max ----
  float m = -INFINITY;
  for (int i = tid; i < C4; i += BLK) {
    v4f v = lrow[i];
    m = fmaxf(m, fmaxf(fmaxf(v.x, v.y), fmaxf(v.z, v.w)));
  }
  m = wave_reduce_max(m);
  if (lane == 0) s_part[wid] = m;
  __syncthreads();
  if (tid == 0) {
    float mm = s_part[0];
#pragma unroll
    for (int w = 1; w < NWAVES; ++w) mm = fmaxf(mm, s_part[w]);
    s_bval = mm;
  }
  __syncthreads();
  m = s_bval;

  // ---- Pass B: e = exp(x - m + eps); overwrite LDS with e; row sum ----
  float acc = 0.f;
  for (int i = tid; i < C4; i += BLK) {
    v4f v = lrow[i];
    v.x = expf(v.x - m + EPS_F);
    v.y = expf(v.y - m + EPS_F);
    v.z = expf(v.z - m + EPS_F);
    v.w = expf(v.w - m + EPS_F);
    acc += (v.x + v.y) + (v.z + v.w);
    lrow[i] = v;
  }
  acc = wave_reduce_sum(acc);
  if (lane == 0) s_part[wid] = acc;
  __syncthreads();                          // also guards s_bval reuse
  if (tid == 0) {
    float ss = 0.f;
#pragma unroll
    for (int w = 0; w < NWAVES; ++w) ss += s_part[w];
    s_bval = ss;
  }
  __syncthreads();
  const float s = s_bval;

  // ---- Pass C: out = e*k / (e*k + (s - e) + eps), non-temporal stores ----
  for (int i = tid; i < C4; i += BLK) {
    v4f e4 = lrow[i];
    v4f r;
    float n;
    n = e4.x * kshift; r.x = n / (n + (s - e4.x) + EPS_F);
    n = e4.y * kshift; r.y = n / (n + (s - e4.y) + EPS_F);
    n = e4.z * kshift; r.z = n / (n + (s - e4.z) + EPS_F);
    n = e4.w * kshift; r.w = n / (n + (s - e4.w) + EPS_F);
    __builtin_nontemporal_store(r, orow + i);
  }
}

extern "C" void kernel_launch(void* const* d_in, const int* in_sizes, int n_in,
                              void* d_out, int out_size, void* d_ws, size_t ws_size,
                              hipStream_t stream) {
  const float* logits = (const float*)d_in[0];
  const float* dist   = (const float*)d_in[1];
  float*       out    = (float*)d_out;

  const int C    = COLS;
  const int rows = in_sizes[0] / C;
  const size_t shbytes = (size_t)C * sizeof(float);   // 128000 B dynamic LDS

  // Allow >64KB dynamic LDS (gfx1250 WGP has 320KB). Idempotent, capture-safe.
  (void)hipFuncSetAttribute(reinterpret_cast<const void*>(&sm_softmax_kernel),
                            hipFuncAttributeMaxDynamicSharedMemorySize, (int)shbytes);

  sm_softmax_kernel<<<rows, BLK, shbytes, stream>>>(logits, dist, out, C);
}